// ConstrainedOutputLayer_47957604827428
// MI455X (gfx1250) — compile-verified
//
#include <hip/hip_runtime.h>
#include <math.h>

// ---------------------------------------------------------------------------
// Problem constants (Galaxy Zoo 2 constrained output layer)
// ---------------------------------------------------------------------------
#define BROWS 32768
#define DDIM  2048
#define NCLS  11
#define NCOL  37
#define NPAD  48          // 3 WMMA N-tiles of 16

typedef __attribute__((ext_vector_type(2))) float v2f;
typedef __attribute__((ext_vector_type(8))) float v8f;

// per-class column start offsets in the concatenated 37-wide output (+ sentinel)
__constant__ int kStart[NCLS + 1] = {0, 3, 5, 7, 9, 13, 15, 18, 25, 28, 31, 37};

struct WPtrs {
    const float* W[NCLS];
    const float* b[NCLS];
};

// ---------------------------------------------------------------------------
// Kernel 1: pack 11 weight matrices -> W_cat[48][2048] (rows 37..47 zero),
//           11 bias vectors        -> b_cat[48]
// ---------------------------------------------------------------------------
__global__ void gz2_pack_kernel(WPtrs p, float* __restrict__ wcat,
                                float* __restrict__ bcat) {
    int idx = blockIdx.x * blockDim.x + threadIdx.x;
    if (idx < NPAD * DDIM) {
        int n = idx / DDIM;
        int k = idx - n * DDIM;
        float val = 0.0f;
        if (n < NCOL) {
            // find owning class
            #pragma unroll
            for (int c = 0; c < NCLS; ++c) {
                if (n >= kStart[c] && n < kStart[c + 1]) {
                    val = p.W[c][(n - kStart[c]) * DDIM + k];
                }
            }
        }
        wcat[idx] = val;
    }
    if (idx < NPAD) {
        float bv = 0.0f;
        if (idx < NCOL) {
            #pragma unroll
            for (int c = 0; c < NCLS; ++c) {
                if (idx >= kStart[c] && idx < kStart[c + 1]) {
                    bv = p.b[c][idx - kStart[c]];
                }
            }
        }
        bcat[idx] = bv;
    }
}

// ---------------------------------------------------------------------------
// Kernel 2: skinny GEMM via V_WMMA_F32_16X16X4_F32 + hierarchy epilogue.
// 4 waves / block, each wave owns one 16-row M tile, all 48 (padded) columns.
// ---------------------------------------------------------------------------
#define WAVES_PER_BLOCK 4
#define LDS_STRIDE 49   // 48 cols + 1 pad to avoid bank conflicts

__device__ __forceinline__ float sigmoidf_(float v) {
    return 1.0f / (1.0f + __expf(-v));
}

__global__ __launch_bounds__(WAVES_PER_BLOCK * 32)
void gz2_gemm_kernel(const float* __restrict__ x,
                     const float* __restrict__ wcat,
                     const float* __restrict__ bcat,
                     float* __restrict__ out) {
    __shared__ float lds[WAVES_PER_BLOCK][16][LDS_STRIDE];

    const int lane  = threadIdx.x & 31;
    const int wave  = threadIdx.x >> 5;
    const int mbase = (blockIdx.x * WAVES_PER_BLOCK + wave) * 16;
    const int mrow  = lane & 15;          // A fragment: M = lane % 16
    const int kk    = (lane >> 4) << 1;   // K pair: lanes 0-15 -> K{0,1}, 16-31 -> K{2,3}

    // A fragment base: x[mbase + mrow][kk + k]
    const float* ap  = x    + (size_t)(mbase + mrow) * DDIM + kk;
    // B fragments: W_cat[ntile*16 + (lane%16)][kk + k]  (B[k][n] = W_cat[n][k])
    const float* bp0 = wcat + (size_t)mrow * DDIM + kk;
    const float* bp1 = bp0 + (size_t)16 * DDIM;
    const float* bp2 = bp0 + (size_t)32 * DDIM;

    v8f c0 = {}, c1 = {}, c2 = {};

    for (int k0 = 0; k0 < DDIM; k0 += 32) {
        // prefetch the x stream ~4KB ahead (L2 speculative)
        if (k0 + 1024 < DDIM) {
            __builtin_prefetch(ap + k0 + 1024, 0, 0);
        }
        #pragma unroll
        for (int ku = 0; ku < 32; ku += 4) {
            const int k = k0 + ku;
            v2f a  = *(const v2f*)(ap  + k);
            v2f b0 = *(const v2f*)(bp0 + k);
            v2f b1 = *(const v2f*)(bp1 + k);
            v2f b2 = *(const v2f*)(bp2 + k);
            c0 = __builtin_amdgcn_wmma_f32_16x16x4_f32(false, a, false, b0,
                                                       (short)0, c0, false, false);
            c1 = __builtin_amdgcn_wmma_f32_16x16x4_f32(false, a, false, b1,
                                                       (short)0, c1, false, false);
            c2 = __builtin_amdgcn_wmma_f32_16x16x4_f32(false, a, false, b2,
                                                       (short)0, c2, false, false);
        }
    }

    // Scatter C fragments to LDS: VGPR v of frag f holds (M = v + 8*(lane/16),
    // N = f*16 + lane%16).
    const int mhalf = (lane >> 4) << 3;
    #pragma unroll
    for (int v = 0; v < 8; ++v) {
        const int m = v + mhalf;
        lds[wave][m][mrow]      = c0[v];
        lds[wave][m][16 + mrow] = c1[v];
        lds[wave][m][32 + mrow] = c2[v];
    }
    __syncthreads();

    // ---------------- hierarchy epilogue: one lane per row -----------------
    if (lane < 16) {
        const int r = lane;
        float lg[NCOL];
        #pragma unroll
        for (int j = 0; j < NCOL; ++j) lg[j] = lds[wave][r][j] + bcat[j];

        float o[NCOL];

        // class1 (cols 0-2): sigmoid
        o[0] = sigmoidf_(lg[0]);
        o[1] = sigmoidf_(lg[1]);
        o[2] = sigmoidf_(lg[2]);
        // class6 (cols 13-14): sigmoid
        o[13] = sigmoidf_(lg[13]);
        o[14] = sigmoidf_(lg[14]);

        auto softmax_scaled = [&](int s, int n, float joint) {
            joint = fminf(joint, 1.0f);
            float mx = lg[s];
            for (int i = 1; i < n; ++i) mx = fmaxf(mx, lg[s + i]);
            float sum = 0.0f;
            for (int i = 0; i < n; ++i) {
                float e = __expf(lg[s + i] - mx);
                o[s + i] = e;
                sum += e;
            }
            float scl = joint / sum;
            for (int i = 0; i < n; ++i) o[s + i] *= scl;
        };

        softmax_scaled(3, 2, o[1]);    // class2  <- class1.2
        softmax_scaled(5, 2, o[4]);    // class3  <- class2.2
        softmax_scaled(7, 2, o[4]);    // class4  <- class2.2
        softmax_scaled(9, 4, o[4]);    // class5  <- class2.2
        softmax_scaled(15, 3, o[0]);   // class7  <- class1.1
        softmax_scaled(18, 7, o[13]);  // class8  <- class6.1
        softmax_scaled(25, 3, o[3]);   // class9  <- class2.1
        softmax_scaled(28, 3, o[7]);   // class10 <- class4.1
        softmax_scaled(31, 6, o[7]);   // class11 <- class4.1

        float* op = out + (size_t)(mbase + r) * NCOL;
        #pragma unroll
        for (int j = 0; j < NCOL; ++j) op[j] = o[j];
    }
}

// ---------------------------------------------------------------------------
// Host entry
// ---------------------------------------------------------------------------
extern "C" void kernel_launch(void* const* d_in, const int* in_sizes, int n_in,
                              void* d_out, int out_size, void* d_ws, size_t ws_size,
                              hipStream_t stream) {
    const float* x = (const float*)d_in[0];
    WPtrs p;
    for (int i = 0; i < NCLS; ++i) {
        p.W[i] = (const float*)d_in[1 + i];
        p.b[i] = (const float*)d_in[1 + NCLS + i];
    }

    float* wcat = (float*)d_ws;                 // 48*2048 floats = 384 KB
    float* bcat = wcat + NPAD * DDIM;           // 48 floats

    {
        int total = NPAD * DDIM;
        int blk = 256;
        gz2_pack_kernel<<<(total + blk - 1) / blk, blk, 0, stream>>>(p, wcat, bcat);
    }
    {
        int mtiles = BROWS / 16;                            // 2048
        int blocks = mtiles / WAVES_PER_BLOCK;              // 512
        gz2_gemm_kernel<<<blocks, WAVES_PER_BLOCK * 32, 0, stream>>>(
            x, wcat, bcat, (float*)d_out);
    }
}